// ComplexMultiHeadAttentionV2_39135742001775
// MI455X (gfx1250) — compile-verified
//
#include <hip/hip_runtime.h>
#include <hip/hip_bf16.h>
#include <math.h>

// ComplexMultiHeadAttentionV2 for MI455X (gfx1250, wave32, WMMA).
//
// Key transform: cos(A-B) expansion turns the bhqkd cos-einsum into a batched
// GEMM  scores = Qc @ Kc^T  with Qc/Kc = [mag*cos(angle) | mag*sin(angle)],
// reducing 134M transcendentals to 0.5M and making every heavy op a GEMM
// that lowers to V_WMMA_F32_16X16X4_F32 (fp32 matches the fp32 reference).
//
// GEMM: 2x2 register tiling per wave (32x32 block) + double-buffered LDS
// staged with GLOBAL_LOAD_ASYNC_TO_LDS_B128 (ASYNCcnt) so the next K-chunk
// streams into LDS while WMMAs consume the current one (no VGPR round-trip).
//
// Workspace (floats): QB | KB | VB | CTX, each 2*8*512*64 = 524288 (8 MB total).

typedef __attribute__((ext_vector_type(2))) float v2f;
typedef __attribute__((ext_vector_type(8))) float v8f;
typedef int b128_t __attribute__((vector_size(16)));  // builtin's element type

#define WMMA_F32(a, b, acc) \
  __builtin_amdgcn_wmma_f32_16x16x4_f32(false, (a), false, (b), (short)0, (acc), false, false)

#define B_DIM 2
#define S_DIM 512
#define D_DIM 512
#define H_DIM 8
#define DK 64
#define DH 32

// ---- CDNA5 async global->LDS copy (ASYNCcnt) ------------------------------
__device__ __forceinline__ void async_copy_b128(const float* g, float* l) {
#if __has_builtin(__builtin_amdgcn_global_load_async_to_lds_b128)
  __builtin_amdgcn_global_load_async_to_lds_b128(
      (__attribute__((address_space(1))) b128_t*)g,
      (__attribute__((address_space(3))) b128_t*)l, 0, 0);
#else
  asm volatile("global_load_async_to_lds_b128 %0, %1, off"
               :
               : "v"((__attribute__((address_space(3))) b128_t*)l), "v"(g)
               : "memory");
#endif
}

__device__ __forceinline__ void wait_async0() {
#if __has_builtin(__builtin_amdgcn_s_wait_asynccnt)
  __builtin_amdgcn_s_wait_asynccnt(0);
#else
  asm volatile("s_wait_asynccnt 0x0" ::: "memory");
#endif
}

// ---------------------------------------------------------------------------
// GEMM: Out = X (M x K) @ W^T (W is N x K row-major).
// Workgroup tile 64(M) x 128(N), 8 waves in 2x4; each wave owns a 32x32 block
// = 2x2 WMMA tiles.  K staged in double-buffered LDS chunks of 32 via async
// loads: prefetch chunk k+1 while computing on chunk k.
// out_mode 0: row-major (M x N).  out_mode 1: head-split (B,H,S,64).
// ---------------------------------------------------------------------------
__global__ __launch_bounds__(256) void gemm_xwt(const float* __restrict__ X,
                                                const float* __restrict__ W,
                                                float* __restrict__ Out,
                                                int K, int N, int out_mode) {
  // Row pad 36 floats: bank = (36*r + c) % 64 -> 16 fragment rows hit 16
  // distinct banks; 36 % 4 == 0 keeps 16B-aligned b128 async stores.
  __shared__ __align__(16) float lX[2][64][36];
  __shared__ __align__(16) float lW[2][128][36];

  const int tid = threadIdx.x;
  const int wave = tid >> 5;
  const int lane = tid & 31;
  const int half = lane >> 4;   // which 16-lane half
  const int ml = lane & 15;     // row (A) / col (B,C) within a 16x16 tile
  const int m0 = blockIdx.x * 64;
  const int n0 = blockIdx.y * 128;
  const int wm = wave >> 2;     // 0..1 : 32-row slab
  const int wn = wave & 3;      // 0..3 : 32-col slab

  const int rA0 = wm * 32 + ml;       // A fragment rows (two M subtiles)
  const int rA1 = rA0 + 16;
  const int rB0 = wn * 32 + ml;       // B fragment rows in W tile (two N subtiles)
  const int rB1 = rB0 + 16;
  const int koff = 2 * half;          // lane<16 -> K=0,1 ; lane>=16 -> K=2,3

  v8f acc00 = {0.f, 0.f, 0.f, 0.f, 0.f, 0.f, 0.f, 0.f};
  v8f acc01 = acc00, acc10 = acc00, acc11 = acc00;

  // async stage of one 32-wide K chunk into LDS buffer `buf`
  auto stage = [&](int k0, int buf) {
#pragma unroll
    for (int u = 0; u < 2; ++u) {  // X: 64x32 = 512 b128 / 256 thr
      const int i = tid + u * 256;
      const int r = i >> 3, c4 = i & 7;
      async_copy_b128(X + (size_t)(m0 + r) * K + k0 + c4 * 4, &lX[buf][r][c4 * 4]);
    }
#pragma unroll
    for (int u = 0; u < 4; ++u) {  // W: 128x32 = 1024 b128 / 256 thr
      const int i = tid + u * 256;
      const int r = i >> 3, c4 = i & 7;
      async_copy_b128(W + (size_t)(n0 + r) * K + k0 + c4 * 4, &lW[buf][r][c4 * 4]);
    }
  };

  stage(0, 0);
  wait_async0();
  __syncthreads();

  for (int k0 = 0; k0 < K; k0 += 32) {
    const int cur = (k0 >> 5) & 1;
    if (k0 + 32 < K) stage(k0 + 32, cur ^ 1);  // prefetch overlaps WMMAs below

    const float(*cx)[36] = lX[cur];
    const float(*cw)[36] = lW[cur];
#pragma unroll
    for (int kk = 0; kk < 32; kk += 4) {
      v2f a0, a1, b0, b1;
      a0.x = cx[rA0][kk + koff + 0];
      a0.y = cx[rA0][kk + koff + 1];
      a1.x = cx[rA1][kk + koff + 0];
      a1.y = cx[rA1][kk + koff + 1];
      b0.x = cw[rB0][kk + koff + 0];
      b0.y = cw[rB0][kk + koff + 1];
      b1.x = cw[rB1][kk + koff + 0];
      b1.y = cw[rB1][kk + koff + 1];
      acc00 = WMMA_F32(a0, b0, acc00);
      acc01 = WMMA_F32(a0, b1, acc01);
      acc10 = WMMA_F32(a1, b0, acc10);
      acc11 = WMMA_F32(a1, b1, acc11);
    }
    // retire prefetch + hand buffers over (all waves done reading `cur`)
    wait_async0();
    __syncthreads();
  }

  // C/D layout: col = lane&15, row = 8*half + vgpr
#pragma unroll
  for (int i = 0; i < 2; ++i) {
#pragma unroll
    for (int j = 0; j < 2; ++j) {
      const v8f acc = (i == 0) ? (j == 0 ? acc00 : acc01)
                               : (j == 0 ? acc10 : acc11);
      const int col = n0 + wn * 32 + j * 16 + ml;
#pragma unroll
      for (int r = 0; r < 8; ++r) {
        const int row = m0 + wm * 32 + i * 16 + 8 * half + r;
        size_t oidx;
        if (out_mode == 0) {
          oidx = (size_t)row * N + col;
        } else {
          const int bb = row >> 9, ss = row & 511;
          const int hh = col >> 6, cc = col & 63;
          oidx = (((((size_t)bb * H_DIM + hh) * S_DIM) + ss) << 6) | (unsigned)cc;
        }
        Out[oidx] = acc[r];
      }
    }
  }
}

// ---------------------------------------------------------------------------
// In-place euler/phase transform on a (B,H,S,64) buffer:
//   re,im = buf[c], buf[c+32];  mag = sqrt(re^2+im^2+1e-9); ph = atan2(im,re)
//   ang  = delta[h,c]*ph (+ bias[h,c]) + s*freqs[c]
//   buf[c] = mag*cos(ang); buf[c+32] = mag*sin(ang)
// One thread owns the (c, c+32) pair -> race-free in place.
// ---------------------------------------------------------------------------
__global__ __launch_bounds__(256) void euler_transform(float* __restrict__ buf,
                                                       const float* __restrict__ delta,
                                                       const float* __restrict__ bias,
                                                       int use_bias) {
  const int idx = blockIdx.x * 256 + threadIdx.x;  // B*H*S*32 total
  const int c = idx & 31;
  const int s = (idx >> 5) & 511;
  const int h = (idx >> 14) & 7;
  const int b = idx >> 17;
  float* row = buf + (((((size_t)b * H_DIM + h) * S_DIM) + s) << 6);
  const float re = row[c];
  const float im = row[c + 32];
  const float mag = sqrtf(re * re + im * im + 1e-9f);
  const float ph = atan2f(im, re);
  const float freq = powf(10000.0f, -(float)c * (2.0f / (float)DK));
  float ang = delta[h * DH + c] * ph + (float)s * freq;
  if (use_bias) ang += bias[h * DH + c];
  float sn, cs;
  sincosf(ang, &sn, &cs);
  row[c] = mag * cs;
  row[c + 32] = mag * sn;
}

// ---------------------------------------------------------------------------
// Fused attention per (b, h, 16-row q tile).  8 waves / 256 threads.
//  phase1: score strip Sc[16][512] = Qc_tile @ Kc^T via WMMA (k-tiles / wave)
//  phase2: row softmax (max, exp, sum) cooperatively; keep exp in Sc,
//          row sums in rsum (normalize at final store)
//  phase3: ctx[16][64] = P @ V via WMMA, K split across wave pairs
// ---------------------------------------------------------------------------
__global__ __launch_bounds__(256) void attn_fused(const float* __restrict__ Qc,
                                                  const float* __restrict__ Kc,
                                                  const float* __restrict__ Vp,
                                                  float* __restrict__ Ctx) {
  __shared__ __align__(16) float Sc[16][516];  // pad: bank = (4r + c) % 64
  __shared__ float red[16][17];
  __shared__ float rsum[16];
  __shared__ float ctxA[16][68];
  __shared__ float ctxB[16][68];

  const int tid = threadIdx.x;
  const int wave = tid >> 5;
  const int lane = tid & 31;
  const int half = lane >> 4;
  const int ml = lane & 15;

  const int bh = blockIdx.x >> 5;          // b*8 + h
  const int q0 = (blockIdx.x & 31) << 4;   // q tile start
  const int b = bh >> 3;
  const int h = bh & 7;

  const float* Qbase = Qc + ((size_t)bh * S_DIM + q0) * DK;
  const float* Kbase = Kc + (size_t)bh * S_DIM * DK;
  const float* Vbase = Vp + (size_t)bh * S_DIM * DK;

  // Preload the A (Q) fragment slices once: row = ml, K elems 4s + 2*half {,+1}
  v2f qf[16];
  {
    const float* qrow = Qbase + (size_t)ml * DK + 2 * half;
#pragma unroll
    for (int s = 0; s < 16; ++s) {
      qf[s].x = qrow[4 * s + 0];
      qf[s].y = qrow[4 * s + 1];
    }
  }

  // ---- phase 1: scores (all branches wave-uniform; EXEC stays all-ones) ----
  for (int kt = wave; kt < 32; kt += 8) {
    const float* krow = Kbase + (size_t)(kt * 16 + ml) * DK + 2 * half;
    v8f acc = {0.f, 0.f, 0.f, 0.f, 0.f, 0.f, 0.f, 0.f};
#pragma unroll
    for (int s = 0; s < 16; ++s) {
      v2f bf;
      bf.x = krow[4 * s + 0];  // B[d][n] = Kc[k0+n][d]; n = ml
      bf.y = krow[4 * s + 1];
      acc = WMMA_F32(qf[s], bf, acc);
    }
#pragma unroll
    for (int r = 0; r < 8; ++r) Sc[8 * half + r][kt * 16 + ml] = acc[r];
  }
  __syncthreads();

  // ---- phase 2: softmax over each of the 16 rows (16 threads per row) ----
  {
    const int row = tid >> 4;
    const int t16 = tid & 15;
    float mx = -INFINITY;
#pragma unroll 4
    for (int j = 0; j < 32; ++j) mx = fmaxf(mx, Sc[row][t16 + 16 * j]);
    red[row][t16] = mx;
    __syncthreads();
    float rmx = -INFINITY;
#pragma unroll
    for (int j = 0; j < 16; ++j) rmx = fmaxf(rmx, red[row][j]);
    float sum = 0.f;
#pragma unroll 4
    for (int j = 0; j < 32; ++j) {
      const float e = __expf(Sc[row][t16 + 16 * j] - rmx);
      Sc[row][t16 + 16 * j] = e;
      sum += e;
    }
    __syncthreads();  // everyone done reading red before reuse
    red[row][t16] = sum;
    __syncthreads();
    if (t16 == 0) {
      float s = 0.f;
#pragma unroll
      for (int j = 0; j < 16; ++j) s += red[row][j];
      rsum[row] = s;
    }
    __syncthreads();
  }

  // ---- phase 3: ctx = P @ V ; wave -> (n-tile, k-half) ----
  {
    const int nt = wave & 3;   // 16-col V tile
    const int kh = wave >> 2;  // 0: k 0..255, 1: k 256..511
    const int c0 = nt * 16;
    v8f acc = {0.f, 0.f, 0.f, 0.f, 0.f, 0.f, 0.f, 0.f};
    const int kbeg = kh * 256;
    for (int k0 = kbeg; k0 < kbeg + 256; k0 += 4) {
      v2f a, bf;
      a.x = Sc[ml][k0 + 2 * half + 0];
      a.y = Sc[ml][k0 + 2 * half + 1];
      const float* vp = Vbase + (size_t)(k0 + 2 * half) * DK + c0 + ml;
      bf.x = vp[0];
      bf.y = vp[DK];  // next K row
      acc = WMMA_F32(a, bf, acc);
    }
    float(*ctxp)[68] = (kh == 0) ? ctxA : ctxB;
#pragma unroll
    for (int r = 0; r < 8; ++r) ctxp[8 * half + r][c0 + ml] = acc[r];
  }
  __syncthreads();

  // ---- combine halves, normalize, store ctx in (B,S,D) layout ----
#pragma unroll
  for (int j = 0; j < 4; ++j) {
    const int idx = tid + 256 * j;  // 16*64 elems
    const int row = idx >> 6;
    const int c = idx & 63;
    const float val = (ctxA[row][c] + ctxB[row][c]) / rsum[row];
    Ctx[((size_t)b * S_DIM + (q0 + row)) * D_DIM + h * DK + c] = val;
  }
}

// ---------------------------------------------------------------------------
extern "C" void kernel_launch(void* const* d_in, const int* in_sizes, int n_in,
                              void* d_out, int out_size, void* d_ws, size_t ws_size,
                              hipStream_t stream) {
  const float* q = (const float*)d_in[0];
  const float* k = (const float*)d_in[1];
  const float* v = (const float*)d_in[2];
  const float* Wq = (const float*)d_in[3];
  const float* Wk = (const float*)d_in[4];
  const float* Wv = (const float*)d_in[5];
  const float* Wo = (const float*)d_in[6];
  const float* delta = (const float*)d_in[7];
  const float* bias = (const float*)d_in[8];
  float* out = (float*)d_out;

  const size_t HBUF = (size_t)B_DIM * H_DIM * S_DIM * DK;  // 524288 floats
  float* QB = (float*)d_ws;
  float* KB = QB + HBUF;
  float* VB = KB + HBUF;
  float* CTX = VB + HBUF;

  const dim3 ggrid(16, 4);  // M=1024/64, N=512/128
  const dim3 gblk(256);

  // Projections into head-split (B,H,S,64) layout.
  gemm_xwt<<<ggrid, gblk, 0, stream>>>(q, Wq, QB, D_DIM, D_DIM, 1);
  gemm_xwt<<<ggrid, gblk, 0, stream>>>(k, Wk, KB, D_DIM, D_DIM, 1);
  gemm_xwt<<<ggrid, gblk, 0, stream>>>(v, Wv, VB, D_DIM, D_DIM, 1);

  // In-place mag*cos / mag*sin phase transform (Q gets bias term).
  euler_transform<<<1024, 256, 0, stream>>>(QB, delta, bias, 1);
  euler_transform<<<1024, 256, 0, stream>>>(KB, delta, bias, 0);

  // Fused scores + softmax + P@V.
  attn_fused<<<B_DIM * H_DIM * (S_DIM / 16), 256, 0, stream>>>(QB, KB, VB, CTX);

  // Output projection.
  gemm_xwt<<<ggrid, gblk, 0, stream>>>(CTX, Wo, out, D_DIM, D_DIM, 0);
}